// PiecewiseChebyshevSeries_4922032521416
// MI455X (gfx1250) — compile-verified
//
#include <hip/hip_runtime.h>

// Problem constants (from reference): B=16, X=256, Y=32, N=131072
#define BB   16
#define XX   256
#define YY   32
#define NN   131072
#define TPB  256
#define IPT  8                 // elements per thread
#define ROW_STRIDE 36          // 32 coeff dwords + 4 pad dwords (bank-conflict spread)
#define SMEM_DW (XX * ROW_STRIDE)

typedef unsigned int v4u __attribute__((ext_vector_type(4)));
typedef int          v8i __attribute__((ext_vector_type(8)));
typedef int          v4i __attribute__((ext_vector_type(4)));

__global__ __launch_bounds__(TPB)
void cheb_piecewise_kernel(const float* __restrict__ z,
                           const float* __restrict__ cheb,
                           float* __restrict__ out) {
    __shared__ __align__(16) float smem[SMEM_DW];   // 36 KB

    const int b = blockIdx.y;
    const float* cheb_b = cheb + (size_t)b * (XX * YY);

#if __has_builtin(__builtin_amdgcn_tensor_load_to_lds) && __has_builtin(__builtin_amdgcn_s_wait_tensorcnt)
    // ---- Tensor Data Mover: DMA 32KB cheb[b] slice -> LDS with row padding ----
    if (threadIdx.x < 32) {   // one wave issues the TDM op (EXEC ignored by TDM)
        unsigned long long ga = (unsigned long long)(const void*)cheb_b;
        unsigned int lds_off = (unsigned int)(unsigned long long)(const void*)&smem[0]; // low 32b = LDS byte addr

        v4u g0;
        g0[0] = 1u;                                   // count=1, is_restore=0, gather off
        g0[1] = lds_off;                              // lds_addr
        g0[2] = (unsigned int)ga;                     // global_addr[31:0]
        g0[3] = ((unsigned int)(ga >> 32) & 0x01FFFFFFu) | (2u << 30); // addr[56:32], type=2 (image)

        v8i g1;
        // data_size=2 (4B) | pad_enable | pad_interval=4 (32 dwords) | pad_amount=3 (4 dwords)
        g1[0] = (2 << 16) | (1 << 20) | (4 << 22) | (3 << 25);
        g1[1] = (int)((8192u & 0xFFFFu) << 16);       // tensor_dim0[15:0] @ bits 63:48
        g1[2] = (int)((8192u >> 16) | (1u << 16));    // tensor_dim0[31:16] | tensor_dim1=1
        g1[3] = (int)(8192u << 16);                   // tile_dim0 = 8192 elements (1-D tile)
        g1[4] = 0;                                    // tile_dim1=0 (unused), tile_dim2=0
        g1[5] = 8192;                                 // tensor_dim0_stride[31:0]
        g1[6] = 0;
        g1[7] = 0;

        v4i g2 = {0, 0, 0, 0};
        v4i g3 = {0, 0, 0, 0};
        v8i g4 = {0, 0, 0, 0, 0, 0, 0, 0};            // extra group in 6-arg clang-23 form

        __builtin_amdgcn_tensor_load_to_lds(g0, g1, g2, g3, g4, /*cpol=*/0);
        __builtin_amdgcn_s_wait_tensorcnt(0);
    }
#else
    // ---- Fallback: cooperative vector load + padded LDS store ----
    for (int i = threadIdx.x; i < XX * (YY / 4); i += TPB) {
        int row = i / (YY / 4);
        int c4  = i % (YY / 4);
        const float4 v = reinterpret_cast<const float4*>(cheb_b)[row * (YY / 4) + c4];
        float* dst = &smem[row * ROW_STRIDE + c4 * 4];
        dst[0] = v.x; dst[1] = v.y; dst[2] = v.z; dst[3] = v.w;
    }
#endif
    __syncthreads();

    const float* zb = z   + (size_t)b * NN;
    float*       ob = out + (size_t)b * NN;
    const int n0 = blockIdx.x * (TPB * IPT) + threadIdx.x;

#pragma unroll
    for (int i = 0; i < IPT; ++i) {
        const int n = n0 + i * TPB;
        const float zv = zb[n];

        // piece index + local disc coordinate
        const float t  = zv + 1.0f;                 // z - a, a = -1
        const float xf = floorf(t * 0.5f);          // divmod(t, 2)
        int xi = (int)xf;
        xi = xi < 0 ? 0 : (xi > XX - 1 ? XX - 1 : xi);
        const float y  = t - 2.0f * xf - 1.0f;      // in [-1, 1)
        const float y2 = y + y;

        const float4* row = reinterpret_cast<const float4*>(&smem[xi * ROW_STRIDE]);

        // cos(k*arccos(y)) == T_k(y): three-term recurrence, two independent FMA chains
        const float4 c0 = row[0];
        float acc = fmaf(c0.y, y, c0.x);            // k=0,1
        float Tm2 = y;                              // will hold T_{k-2} after first steps
        float Ta  = fmaf(y2, y, -1.0f);             // T2
        acc = fmaf(c0.z, Ta, acc);
        float Tm1 = fmaf(y2, Ta, -y);               // T3
        acc = fmaf(c0.w, Tm1, acc);
        Tm2 = Ta;

#pragma unroll
        for (int chunk = 1; chunk < YY / 4; ++chunk) {
            const float4 c = row[chunk];
            const float t0 = fmaf(y2, Tm1, -Tm2); acc = fmaf(c.x, t0, acc);
            const float t1 = fmaf(y2, t0,  -Tm1); acc = fmaf(c.y, t1, acc);
            const float t2 = fmaf(y2, t1,  -t0);  acc = fmaf(c.z, t2, acc);
            const float t3 = fmaf(y2, t2,  -t1);  acc = fmaf(c.w, t3, acc);
            Tm2 = t2; Tm1 = t3;
        }

        ob[n] = acc;
    }
}

extern "C" void kernel_launch(void* const* d_in, const int* in_sizes, int n_in,
                              void* d_out, int out_size, void* d_ws, size_t ws_size,
                              hipStream_t stream) {
    const float* z    = (const float*)d_in[0];   // (B, N)
    const float* cheb = (const float*)d_in[1];   // (B, X, Y)
    float*       out  = (float*)d_out;           // (B, N)

    dim3 grid(NN / (TPB * IPT), BB);             // (64, 16)
    dim3 block(TPB);
    cheb_piecewise_kernel<<<grid, block, 0, stream>>>(z, cheb, out);
}